// Elman_57836029608484
// MI455X (gfx1250) — compile-verified
//
#include <hip/hip_runtime.h>
#include <hip/hip_bf16.h>

// ---------------------------------------------------------------------------
// Elman RNN forward for MI455X (gfx1250, wave32, WMMA, async-LDS, TDM).
//   Phase 1: U[t][b][h] = b1[h] + emb[x[b,t]] @ W1[:, :E].T   (parallel GEMM)
//   Phase 2: h_t = tanh(U_t + h_{t-1} @ W1[:, E:].T)          (persistent, grid-sync)
//            pooled = max_t h_t   (in registers)
//   Phase 3: logits = pooled @ W2.T + b2
// Output: [logits (64*2) | hT (64*512)] fp32.
// ---------------------------------------------------------------------------

typedef float v2f __attribute__((ext_vector_type(2)));
typedef float v8f __attribute__((ext_vector_type(8)));
typedef unsigned int v4u __attribute__((ext_vector_type(4)));
typedef int v4i __attribute__((ext_vector_type(4)));
typedef int v8i __attribute__((ext_vector_type(8)));

#define BB   64
#define TT   512
#define EE   512
#define HH   512
#define NOUT 2
#define REC_BLOCKS 8
// LDS row stride for 512-float rows: pad 2 DWORDs per 256 DWORDs (matches TDM
// pad_interval=7/pad_amount=1) -> 516 floats, 2064 B => 4-bank rotation/row.
#define HS   516
#define US   68    // padded stride for the 64-float U rows

#ifndef __has_builtin
#define __has_builtin(x) 0
#endif
#if __has_builtin(__builtin_amdgcn_global_load_async_to_lds_b128)
#define HAVE_ASYNC_LDS 1
#else
#define HAVE_ASYNC_LDS 0
#endif
#if __has_builtin(__builtin_amdgcn_tensor_load_to_lds) && \
    __has_builtin(__builtin_amdgcn_s_wait_tensorcnt)
#define HAVE_TDM 1
#else
#define HAVE_TDM 0
#endif

// AS(1) prints as `__device__`, AS(3) as `__shared__` in HIP clang diagnostics.
typedef __attribute__((address_space(1))) v4i gv4i;   // global int4
typedef __attribute__((address_space(3))) v4i lv4i;   // LDS int4
typedef __attribute__((address_space(3))) void lds_t; // LDS void (offset extraction)

__device__ __forceinline__ v8f wmma_f32(v2f a, v2f b, v8f c) {
  // 8 args: (neg_a, A, neg_b, B, c_mod, C, reuse_a, reuse_b)
  return __builtin_amdgcn_wmma_f32_16x16x4_f32(false, a, false, b, (short)0, c,
                                               false, false);
}

// 16-byte global->LDS copy: async (ASYNCcnt) on CDNA5, VGPR bounce otherwise.
__device__ __forceinline__ void copy16_g2l(const float* g, float* l) {
#if HAVE_ASYNC_LDS
  __builtin_amdgcn_global_load_async_to_lds_b128((gv4i*)g, (lv4i*)l, 0, 0);
#else
  *(float4*)l = *(const float4*)g;
#endif
}
__device__ __forceinline__ void wait_async() {
#if HAVE_ASYNC_LDS
  asm volatile("s_wait_asynccnt 0x0" ::: "memory");
#endif
}

// ---- grid-wide barrier (persistent kernel, REC_BLOCKS co-resident blocks) ----
__device__ __forceinline__ void gridBarrier(int* cnt, int* gen, int nblocks) {
  __syncthreads();
  if (threadIdx.x == 0) {
    __threadfence();
    int g = __hip_atomic_load(gen, __ATOMIC_RELAXED, __HIP_MEMORY_SCOPE_AGENT);
    int prev = __hip_atomic_fetch_add(cnt, 1, __ATOMIC_ACQ_REL, __HIP_MEMORY_SCOPE_AGENT);
    if (prev == nblocks - 1) {
      __hip_atomic_store(cnt, 0, __ATOMIC_RELAXED, __HIP_MEMORY_SCOPE_AGENT);
      __hip_atomic_fetch_add(gen, 1, __ATOMIC_RELEASE, __HIP_MEMORY_SCOPE_AGENT);
    } else {
      while (__hip_atomic_load(gen, __ATOMIC_ACQUIRE, __HIP_MEMORY_SCOPE_AGENT) == g) {
        __builtin_amdgcn_s_sleep(2);
      }
    }
    __threadfence();
  }
  __syncthreads();
}

// ---------------------------------------------------------------------------
// Phase 1: embedding gather + x-projection.  Grid (1024, 8), 256 thr (8 waves).
// Each wave owns one 16x16 tile; block covers 32 rows x 64 cols. Writes U
// transposed to [T][B][H] so phase 2 reads contiguous per-t slabs.
// ---------------------------------------------------------------------------
__global__ __launch_bounds__(256) void k_embed_xproj(
    const int* __restrict__ x, const float* __restrict__ emb,
    const float* __restrict__ W1, const float* __restrict__ b1,
    float* __restrict__ Ut) {
  const int tid  = threadIdx.x;
  const int lane = tid & 31;
  const int wave = tid >> 5;
  const int mt = wave & 1;
  const int nt = wave >> 1;
  const int row0 = blockIdx.x * 32 + mt * 16;   // bt row base
  const int col0 = blockIdx.y * 64 + nt * 16;   // hidden col base
  const int lm = lane & 15;
  const int kh = (lane >> 4) << 1;   // K sub-offset per 16x4 f32 A/B layout
  const int mhalf = (lane >> 4) << 3;

  const int tok = x[row0 + lm];
  const float* embRow = emb + (long)tok * EE;
  const float* wRow   = W1  + (long)(col0 + lm) * (2 * EE);  // x-part: cols [0,E)

  const float bias = b1[col0 + lm];
  v8f c;
#pragma unroll
  for (int r = 0; r < 8; ++r) c[r] = bias;

#pragma unroll 4
  for (int k0 = 0; k0 < EE; k0 += 4) {
    v2f a = *(const v2f*)(embRow + k0 + kh);
    v2f b = *(const v2f*)(wRow   + k0 + kh);
    c = wmma_f32(a, b, c);
  }

  const int colc = col0 + lm;
#pragma unroll
  for (int r = 0; r < 8; ++r) {
    int bt = row0 + r + mhalf;
    int b_ = bt >> 9;
    int t_ = bt & (TT - 1);
    Ut[((long)t_ * BB + b_) * HH + colc] = c[r];
  }
}

// ---------------------------------------------------------------------------
// Phase 2: recurrence. 8 persistent blocks x 256 threads; block b owns hidden
// cols [64b, 64b+64). LDS: W-chunk [64][HS] (TDM) + h staging [64][HS] (async)
// + U tile [64][US] (async) = ~275 KB of CDNA5's 320 KB WGP LDS.
// ---------------------------------------------------------------------------
__global__ __launch_bounds__(256) void k_recurrent(
    const float* __restrict__ Ut, const float* __restrict__ W1,
    float* __restrict__ hbuf0, float* __restrict__ hbuf1,
    float* __restrict__ pooled, float* __restrict__ hT_out,
    int* __restrict__ bar) {
  extern __shared__ float lds[];
  float* Wl = lds;                      // [64][HS] recurrent-weight chunk
  float* hl = lds + 64 * HS;            // [64][HS] h_{t-1} staging
  float* Ul = lds + 2 * 64 * HS;        // [64][US] U tile for current t

  const int tid   = threadIdx.x;
  const int lane  = tid & 31;
  const int wave  = tid >> 5;
  const int mt    = wave & 3;
  const int npair = wave >> 2;
  const int colbase = blockIdx.x * 64;
  const int lm = lane & 15;
  const int kh = (lane >> 4) << 1;
  const int mhalf = (lane >> 4) << 3;

  // ---- One-time load of recurrent weight chunk W1[colbase+n, E+k] into LDS.
#if HAVE_TDM
  if (wave == 0) {
    // D# group0: count=1 | lds_addr | 57-bit global addr | type=2
    unsigned long long ga =
        (unsigned long long)(const void*)(W1 + (long)colbase * (2 * EE) + EE);
    unsigned int ldsoff = (unsigned int)(unsigned long long)(lds_t*)(void*)Wl;
    v4u g0 = {1u, ldsoff, (unsigned int)ga,
              (unsigned int)(((ga >> 32) & 0x01FFFFFFu) | (2u << 30))};
    v8i g1;
    g1[0] = (2 << 16)       // data_size = 4 B
          | (1 << 20)       // pad_enable
          | (7 << 22)       // pad_interval = 256 DWORDs
          | (1 << 25);      // pad_amount   = 2 DWORDs  -> row stride 516 floats
    g1[1] = (int)(512u << 16);   // tensor_dim0 = 512  (bits [79:48])
    g1[2] = (int)(64u << 16);    // tensor_dim1 = 64   (bits [111:80])
    g1[3] = (int)(512u << 16);   // tile_dim0   = 512  (bits [127:112])
    g1[4] = 64;                  // tile_dim1   = 64   (bits [143:128])
    g1[5] = 1024;                // tensor_dim0_stride = 1024 elts (skip x-part)
    g1[6] = 0;
    g1[7] = 0;
    v4i gz4 = {0, 0, 0, 0};
    v8i gz8 = {0, 0, 0, 0, 0, 0, 0, 0};
    __builtin_amdgcn_tensor_load_to_lds(g0, g1, gz4, gz4, gz8, 0);
    __builtin_amdgcn_s_wait_tensorcnt(0);
  }
#else
  for (int v = tid; v < 64 * (HH / 4); v += 256) {
    int n  = v >> 7;
    int k4 = (v & 127) << 2;
    float4 w = *(const float4*)(W1 + (long)(colbase + n) * (2 * EE) + EE + k4);
    *(float4*)(Wl + n * HS + k4 + ((k4 >> 8) << 1)) = w;
  }
#endif
  __syncthreads();

  v8f hmax[2];
#pragma unroll
  for (int j = 0; j < 2; ++j)
#pragma unroll
    for (int r = 0; r < 8; ++r) hmax[j][r] = -1e30f;

  for (int t = 0; t < TT; ++t) {
    // ---- async-stage U tile [64 x 64] for this timestep into LDS
    for (int v = tid; v < 64 * 16; v += 256) {
      int row = v >> 4;
      int q   = (v & 15) << 2;
      copy16_g2l(Ut + ((long)t * BB + row) * HH + colbase + q,
                 Ul + row * US + q);
    }
    // ---- async-stage h_{t-1} [64 x 512] (zeros at t==0)
    if (t == 0) {
      for (int v = tid; v < 64 * (HS / 4); v += 256)
        *(float4*)(hl + (v << 2)) = make_float4(0.f, 0.f, 0.f, 0.f);
    } else {
      const float* hsrc = ((t & 1) == 1) ? hbuf0 : hbuf1;
      for (int v = tid; v < BB * (HH / 4); v += 256) {
        int b_ = v >> 7;
        int k4 = (v & 127) << 2;
        copy16_g2l(hsrc + b_ * HH + k4, hl + b_ * HS + k4 + ((k4 >> 8) << 1));
      }
    }
    wait_async();
    __syncthreads();

    // Prefetch next timestep's U slab into L2 while we compute this one.
    if (t + 1 < TT)
      __builtin_prefetch(Ut + ((long)(t + 1) * BB + (tid >> 3)) * HH + colbase
                            + ((tid & 7) << 3), 0, 1);

    // Accumulators initialized from U[t] (bias + x-projection already folded).
    v8f c[2];
#pragma unroll
    for (int j = 0; j < 2; ++j) {
      const int cc = (npair * 2 + j) * 16 + lm;
#pragma unroll
      for (int r = 0; r < 8; ++r) {
        int b_ = mt * 16 + r + mhalf;
        c[j][r] = Ul[b_ * US + cc];
      }
    }

    const float* arow  = hl + (mt * 16 + lm) * HS;
    const float* brow0 = Wl + ((npair * 2 + 0) * 16 + lm) * HS;
    const float* brow1 = Wl + ((npair * 2 + 1) * 16 + lm) * HS;
#pragma unroll 4
    for (int k0 = 0; k0 < HH; k0 += 4) {
      const int ko = k0 + kh + ((k0 >> 8) << 1);   // skip mid-row pad
      v2f a   = *(const v2f*)(arow  + ko);
      v2f b0  = *(const v2f*)(brow0 + ko);
      v2f b1v = *(const v2f*)(brow1 + ko);
      c[0] = wmma_f32(a, b0,  c[0]);
      c[1] = wmma_f32(a, b1v, c[1]);
    }

    float* hdst = ((t & 1) == 0) ? hbuf0 : hbuf1;
#pragma unroll
    for (int j = 0; j < 2; ++j) {
      const int col = colbase + (npair * 2 + j) * 16 + lm;
#pragma unroll
      for (int r = 0; r < 8; ++r) {
        int b_ = mt * 16 + r + mhalf;
        float v = tanhf(c[j][r]);
        hmax[j][r] = fmaxf(hmax[j][r], v);
        hdst[b_ * HH + col] = v;
      }
    }
    gridBarrier(&bar[0], &bar[1], REC_BLOCKS);
  }

  // pooled = max_t h_t (each (b,col) owned by exactly one lane/reg)
#pragma unroll
  for (int j = 0; j < 2; ++j) {
    const int col = colbase + (npair * 2 + j) * 16 + lm;
#pragma unroll
    for (int r = 0; r < 8; ++r) {
      int b_ = mt * 16 + r + mhalf;
      pooled[b_ * HH + col] = hmax[j][r];
    }
  }
  // hT lives in hbuf1 (t = 511 is odd); copy this block's 1/8 share.
  const float* hfin = hbuf1;
  const int base = blockIdx.x * (BB * HH / REC_BLOCKS);
  for (int i = tid; i < BB * HH / REC_BLOCKS; i += 256)
    hT_out[base + i] = hfin[base + i];
}

// ---------------------------------------------------------------------------
// Phase 3: logits = pooled @ W2.T + b2   (64x2, trivial GEMV)
// ---------------------------------------------------------------------------
__global__ __launch_bounds__(128) void k_logits(
    const float* __restrict__ pooled, const float* __restrict__ W2,
    const float* __restrict__ b2, float* __restrict__ out) {
  const int tid = threadIdx.x;
  const int o = tid & 1, b_ = tid >> 1;
  float acc = b2[o];
  const float* p = pooled + b_ * HH;
  const float* w = W2 + o * HH;
#pragma unroll 8
  for (int h = 0; h < HH; ++h) acc += p[h] * w[h];
  out[b_ * NOUT + o] = acc;
}

// Fallback scratch if d_ws is too small (zero-init .bss in the code object).
#define WS_FLOATS ((size_t)TT * BB * HH + 3u * BB * HH + 64u)
__device__ float g_fallback_ws[WS_FLOATS];

extern "C" void kernel_launch(void* const* d_in, const int* in_sizes, int n_in,
                              void* d_out, int out_size, void* d_ws, size_t ws_size,
                              hipStream_t stream) {
  const int*   x   = (const int*)d_in[0];
  const float* emb = (const float*)d_in[1];
  const float* W1  = (const float*)d_in[2];
  const float* b1  = (const float*)d_in[3];
  const float* W2  = (const float*)d_in[4];
  const float* b2  = (const float*)d_in[5];
  float* out = (float*)d_out;

  float* base = (float*)d_ws;
  if (ws_size < WS_FLOATS * sizeof(float)) {
    void* p = 0;
    (void)hipGetSymbolAddress(&p, HIP_SYMBOL(g_fallback_ws));
    base = (float*)p;
  }
  float* Ut     = base;                              // [T][B][H]
  float* hbuf0  = Ut + (size_t)TT * BB * HH;         // [B][H]
  float* hbuf1  = hbuf0 + BB * HH;                   // [B][H]
  float* pooled = hbuf1 + BB * HH;                   // [B][H]
  int*   bar    = (int*)(pooled + BB * HH);          // {count, generation}

  (void)hipMemsetAsync(bar, 0, 2 * sizeof(int), stream);

  dim3 g1(BB * TT / 32, HH / 64);
  k_embed_xproj<<<g1, 256, 0, stream>>>(x, emb, W1, b1, Ut);

  const int ldsBytes = (2 * 64 * HS + 64 * US) * (int)sizeof(float); // ~275 KB
  (void)hipFuncSetAttribute(reinterpret_cast<const void*>(k_recurrent),
                            hipFuncAttributeMaxDynamicSharedMemorySize, ldsBytes);
  k_recurrent<<<REC_BLOCKS, 256, ldsBytes, stream>>>(
      Ut, W1, hbuf0, hbuf1, pooled, out + BB * NOUT, bar);

  k_logits<<<1, 128, 0, stream>>>(pooled, W2, b2, out);
}